// SpectralConv_86371792323181
// MI455X (gfx1250) — compile-verified
//
#include <hip/hip_runtime.h>

typedef __attribute__((ext_vector_type(2))) float v2f;
typedef __attribute__((ext_vector_type(8))) float v8f;

#define D 128            // feature dim (in == out == 128 per reference)
#define GEMM_ROWS 32     // rows of output per block
#define KCHUNK 64        // K-slice of W staged in LDS per phase

// ---------------------------------------------------------------- zero ws
__global__ void sc_zero_kernel(float4* __restrict__ p, int count4) {
    int i = blockIdx.x * blockDim.x + threadIdx.x;
    if (i < count4) p[i] = make_float4(0.f, 0.f, 0.f, 0.f);
}

// ---------------------------------------------------------------- degree
__global__ void sc_degree_kernel(const int* __restrict__ row, float* __restrict__ deg, int E) {
    int e = blockIdx.x * blockDim.x + threadIdx.x;
    if (e < E) {
        __hip_atomic_fetch_add(&deg[row[e]], 1.0f,
                               __ATOMIC_RELAXED, __HIP_MEMORY_SCOPE_AGENT);
    }
}

// ---------------------------------------------------------------- dis = rsqrt(deg+1)
__global__ void sc_rsqrt_kernel(float* __restrict__ dis, int N) {
    int i = blockIdx.x * blockDim.x + threadIdx.x;
    if (i < N) dis[i] = rsqrtf(dis[i] + 1.0f);
}

// ---------------------------------------------------------------- scatter: agg[row] += dis[col]*v[col]
// 32 lanes per edge, lane handles 4 contiguous channels (coalesced float4 gather)
__global__ void sc_scatter_kernel(const float* __restrict__ v,
                                  const int* __restrict__ row,
                                  const int* __restrict__ col,
                                  const float* __restrict__ dis,
                                  float* __restrict__ agg, int E) {
    int t = blockIdx.x * blockDim.x + threadIdx.x;
    int total = E * 32;
    if (t >= total) return;
    int e    = t >> 5;
    int lane = t & 31;
    int r = row[e];
    int c = col[e];
    float s = dis[c];
    float4 vv = *(const float4*)(v + (size_t)c * D + lane * 4);
    float* dst = agg + (size_t)r * D + lane * 4;
    __hip_atomic_fetch_add(dst + 0, s * vv.x, __ATOMIC_RELAXED, __HIP_MEMORY_SCOPE_AGENT);
    __hip_atomic_fetch_add(dst + 1, s * vv.y, __ATOMIC_RELAXED, __HIP_MEMORY_SCOPE_AGENT);
    __hip_atomic_fetch_add(dst + 2, s * vv.z, __ATOMIC_RELAXED, __HIP_MEMORY_SCOPE_AGENT);
    __hip_atomic_fetch_add(dst + 3, s * vv.w, __ATOMIC_RELAXED, __HIP_MEMORY_SCOPE_AGENT);
}

// ---------------------------------------------------------------- Lv = v - dis*(agg + dis*v)  (in-place on agg)
__global__ void sc_finish_kernel(const float* __restrict__ v,
                                 const float* __restrict__ dis,
                                 float* __restrict__ agg, int N) {
    int t = blockIdx.x * blockDim.x + threadIdx.x;
    int total = N * (D / 4);
    if (t >= total) return;
    int node = t / (D / 4);
    float dsc = dis[node];
    float4 xv = *(const float4*)(v + (size_t)t * 4);
    float4 av = *(const float4*)(agg + (size_t)t * 4);
    float4 o;
    o.x = xv.x - dsc * (av.x + dsc * xv.x);
    o.y = xv.y - dsc * (av.y + dsc * xv.y);
    o.z = xv.z - dsc * (av.z + dsc * xv.z);
    o.w = xv.w - dsc * (av.w + dsc * xv.w);
    *(float4*)(agg + (size_t)t * 4) = o;
}

// ---------------------------------------------------------------- fused poly + GEMM (+bias) via f32 WMMA
// out[n, :] = (a*L2x + b*Lx + c*x)[n, :] @ W + bias
__launch_bounds__(256)
__global__ void sc_gemm_wmma_kernel(const float* __restrict__ x,
                                    const float* __restrict__ Lx,
                                    const float* __restrict__ L2x,
                                    const float* __restrict__ W,
                                    const float* __restrict__ bias,
                                    const float* __restrict__ pa,
                                    const float* __restrict__ pb,
                                    const float* __restrict__ pc,
                                    float* __restrict__ out, int N) {
    // A tile: 32 rows x 128 K, stride 132 (bank-conflict-free column reads)
    __shared__ float sA[GEMM_ROWS * 132];
    // W slice: 64 K-rows x 128 N, stride 136 (half-wave reads K and K+2 rows ->
    // 2*136 % 64banks == 16 -> disjoint 16-bank sets)
    __shared__ float sW[KCHUNK * 136];

    const float a = pa[0], b = pb[0], c = pc[0];
    const int row0 = blockIdx.x * GEMM_ROWS;
    const int tid  = threadIdx.x;

    // ---- load A tile with fused polynomial: 32*128 floats = 1024 float4, 4/thread
    for (int i = tid; i < GEMM_ROWS * (D / 4); i += 256) {
        int rr = i >> 5;          // local row
        int q  = i & 31;          // float4 group
        int gr = row0 + rr; if (gr >= N) gr = N - 1;   // clamp (N % 32 == 0 normally)
        size_t g = (size_t)gr * D + q * 4;
        float4 xv = *(const float4*)(x   + g);
        float4 l1 = *(const float4*)(Lx  + g);
        float4 l2 = *(const float4*)(L2x + g);
        float4 p;
        p.x = a * l2.x + b * l1.x + c * xv.x;
        p.y = a * l2.y + b * l1.y + c * xv.y;
        p.z = a * l2.z + b * l1.z + c * xv.z;
        p.w = a * l2.w + b * l1.w + c * xv.w;
        *(float4*)&sA[rr * 132 + q * 4] = p;
    }
    __syncthreads();

    const int wave = tid >> 5;
    const int lane = tid & 31;
    const int half = lane >> 4;   // 0: K+0/K+1 , 1: K+2/K+3 (ISA 16x4 A layout)
    const int l16  = lane & 15;
    const int nb   = wave * 16;   // 16-column slice owned by this wave

    v8f acc0 = {};                // rows row0 .. row0+15
    v8f acc1 = {};                // rows row0+16 .. row0+31

    for (int ph = 0; ph < D / KCHUNK; ++ph) {
        // stage K-slice of W into LDS: 64*128 floats = 2048 float4, 8/thread
        for (int i = tid; i < KCHUNK * (D / 4); i += 256) {
            int rr = i >> 5;
            int q  = i & 31;
            *(float4*)&sW[rr * 136 + q * 4] =
                *(const float4*)(W + (size_t)(ph * KCHUNK + rr) * D + q * 4);
        }
        __syncthreads();

        for (int k0 = 0; k0 < KCHUNK; k0 += 4) {
            const int ka = k0 + 2 * half;          // this half-wave's K pair
            const int kA = ph * KCHUNK + ka;       // absolute K for A tile
            v2f A0, A1, B;
            A0.x = sA[l16 * 132 + kA];
            A0.y = sA[l16 * 132 + kA + 1];
            A1.x = sA[(l16 + 16) * 132 + kA];
            A1.y = sA[(l16 + 16) * 132 + kA + 1];
            B.x  = sW[ka * 136 + nb + l16];
            B.y  = sW[(ka + 1) * 136 + nb + l16];
            // D = A(16x4, f32) x B(4x16, f32) + C(16x16, f32)
            acc0 = __builtin_amdgcn_wmma_f32_16x16x4_f32(
                       false, A0, false, B, (short)0, acc0, false, false);
            acc1 = __builtin_amdgcn_wmma_f32_16x16x4_f32(
                       false, A1, false, B, (short)0, acc1, false, false);
        }
        __syncthreads();   // all reads of sW done before next phase overwrites
    }

    // ---- epilogue: C/D layout -> VGPR r: lanes0-15 M=r, lanes16-31 M=8+r
    const float bv = bias[nb + l16];
    #pragma unroll
    for (int r = 0; r < 8; ++r) {
        int m = r + 8 * half;
        int g0 = row0 + m;
        int g1 = row0 + 16 + m;
        if (g0 < N) out[(size_t)g0 * D + nb + l16] = acc0[r] + bv;
        if (g1 < N) out[(size_t)g1 * D + nb + l16] = acc1[r] + bv;
    }
}

// ================================================================ launcher
extern "C" void kernel_launch(void* const* d_in, const int* in_sizes, int n_in,
                              void* d_out, int out_size, void* d_ws, size_t ws_size,
                              hipStream_t stream) {
    const float* x     = (const float*)d_in[0];
    const int*   ei    = (const int*)d_in[1];
    const float* W     = (const float*)d_in[2];
    const float* bias  = (const float*)d_in[3];
    const float* pa    = (const float*)d_in[4];
    const float* pb    = (const float*)d_in[5];
    const float* pc    = (const float*)d_in[6];
    float*       out   = (float*)d_out;

    const int N = in_sizes[0] / D;       // 100000
    const int E = in_sizes[1] / 2;       // 1600000
    const int* row = ei;
    const int* col = ei + E;

    // workspace layout: [dis (padded)] [Lx N*128] [L2x N*128]
    size_t disFloats = ((size_t)N + 255) & ~(size_t)255;
    float* dis = (float*)d_ws;
    float* Lx  = dis + disFloats;
    float* L2x = Lx + (size_t)N * D;
    size_t totalFloats = disFloats + 2 * (size_t)N * D;

    const int TB = 256;

    // 1) zero dis + Lx + L2x (scatter targets must start from 0 every call)
    int count4 = (int)(totalFloats / 4);
    sc_zero_kernel<<<(count4 + TB - 1) / TB, TB, 0, stream>>>((float4*)d_ws, count4);

    // 2) degree via hardware float atomics, then dis = rsqrt(deg + 1)
    sc_degree_kernel<<<(E + TB - 1) / TB, TB, 0, stream>>>(row, dis, E);
    sc_rsqrt_kernel<<<(N + TB - 1) / TB, TB, 0, stream>>>(dis, N);

    // 3) Lx = L @ x
    int scatterThreads = E * 32;
    sc_scatter_kernel<<<(scatterThreads + TB - 1) / TB, TB, 0, stream>>>(x, row, col, dis, Lx, E);
    int finThreads = N * (D / 4);
    sc_finish_kernel<<<(finThreads + TB - 1) / TB, TB, 0, stream>>>(x, dis, Lx, N);

    // 4) L2x = L @ Lx
    sc_scatter_kernel<<<(scatterThreads + TB - 1) / TB, TB, 0, stream>>>(Lx, row, col, dis, L2x, E);
    sc_finish_kernel<<<(finThreads + TB - 1) / TB, TB, 0, stream>>>(Lx, dis, L2x, N);

    // 5) out = (a*L2x + b*Lx + c*x) @ W + bias   (fp32 WMMA, poly fused into A load)
    int gemmBlocks = (N + GEMM_ROWS - 1) / GEMM_ROWS;
    sc_gemm_wmma_kernel<<<gemmBlocks, TB, 0, stream>>>(x, Lx, L2x, W, bias,
                                                       pa, pb, pc, out, N);
}